// FGateEncoder_72636486910642
// MI455X (gfx1250) — compile-verified
//
#include <hip/hip_runtime.h>
#include <hip/hip_bf16.h>
#include <math.h>

// ---------------------------------------------------------------------------
// FGateEncoder forward for MI455X (gfx1250, wave32).
// fp32 V_WMMA_F32_16X16X4_F32 matmuls; B-panels staged through LDS with
// double-buffered GLOBAL_LOAD_ASYNC_TO_LDS_B128 (ASYNCcnt-throttled).
// ---------------------------------------------------------------------------

typedef __attribute__((ext_vector_type(2))) float v2f;
typedef __attribute__((ext_vector_type(8))) float v8f;

#define Bsz 64
#define Ssz 512
#define Hsz 512
#define G4  2048   // 4*H
#define G2  1024   // 2*H

#define CHUNK    64                  // k-depth per async chunk
#define BSTRIDE  72                  // padded floats per k-row in LDS (bank-safe)
#define BUFWORDS (CHUNK * BSTRIDE)   // 4608 floats per buffer
#define NCHUNK   (Hsz / CHUNK)       // 8

__device__ __forceinline__ float sigmf(float x) { return 1.0f / (1.0f + expf(-x)); }

// ---------------------------------------------------------------------------
// Issue one 64k x 64col B-panel chunk as async global->LDS b128 copies.
// 1024 x 16B transfers / 128 threads = 8 async instructions per wave.
// idx -> k = idx>>4, tile = (idx>>2)&3, quarter = idx&3 (4 floats each).
// ---------------------------------------------------------------------------
__device__ __forceinline__ void fg_async_issue(
    const float* __restrict__ Bbase, int ldB, int colOffset, int colStride,
    unsigned ldsBase, int chunk, int buf, int tid) {
#pragma unroll
    for (int pass = 0; pass < 8; ++pass) {
        int idx  = tid + pass * 128;
        int k    = idx >> 4;
        int tile = (idx >> 2) & 3;
        int q    = idx & 3;
        const float* gp = Bbase + (size_t)(chunk * CHUNK + k) * ldB
                        + colOffset + tile * colStride + q * 4;
        unsigned lds = ldsBase
                     + (unsigned)(buf * BUFWORDS + k * BSTRIDE + tile * 16 + q * 4) * 4u;
        asm volatile("global_load_async_to_lds_b128 %0, %1, off"
                     :: "v"(lds), "v"((unsigned long long)(size_t)gp)
                     : "memory");
    }
}

// ---------------------------------------------------------------------------
// Consume one LDS-resident chunk: 16 WMMA k-steps x 4 n-tiles.
// A: per-lane global row pointer (M = l16); B: from LDS (N = l16).
// ---------------------------------------------------------------------------
__device__ __forceinline__ void fg_consume(
    const float* __restrict__ Ap, int kBase, const float* __restrict__ BsBuf,
    int half, int l16, v8f acc[4]) {
    for (int k2 = 0; k2 < CHUNK; k2 += 4) {
        v2f a = *(const v2f*)(Ap + kBase + k2 + 2 * half);
        const float* br = BsBuf + (k2 + 2 * half) * BSTRIDE + l16;
#pragma unroll
        for (int t = 0; t < 4; ++t) {
            v2f bb;
            bb.x = br[t * 16];
            bb.y = br[BSTRIDE + t * 16];
            acc[t] = __builtin_amdgcn_wmma_f32_16x16x4_f32(false, a, false, bb,
                                                           (short)0, acc[t], false, false);
        }
    }
}

// ---------------------------------------------------------------------------
// Full K=512 pipelined matmul: double-buffered async B-panel + WMMA consume.
// ---------------------------------------------------------------------------
__device__ __forceinline__ void fg_mm_pipe(
    const float* __restrict__ Ap, const float* __restrict__ Bbase, int ldB,
    int colOffset, int colStride, float* BsShared, int tid, v8f acc[4]) {
    const int lane = tid & 31;
    const int half = lane >> 4;
    const int l16  = lane & 15;
    unsigned ldsBase = (unsigned)(size_t)BsShared;

    fg_async_issue(Bbase, ldB, colOffset, colStride, ldsBase, 0, 0, tid);
#pragma unroll 1
    for (int kc = 0; kc < NCHUNK; ++kc) {
        if (kc + 1 < NCHUNK) {
            fg_async_issue(Bbase, ldB, colOffset, colStride, ldsBase,
                           kc + 1, (kc + 1) & 1, tid);
            // chunk kc done when <= 8 (the just-issued chunk) remain outstanding
            asm volatile("s_wait_asynccnt 0x8" ::: "memory");
        } else {
            asm volatile("s_wait_asynccnt 0x0" ::: "memory");
        }
        __syncthreads();   // publish chunk kc to all waves
        fg_consume(Ap, kc * CHUNK, BsShared + (kc & 1) * BUFWORDS, half, l16, acc);
        __syncthreads();   // everyone done with buf (kc&1) before it is overwritten
    }
}

// ---------------------------------------------------------------------------
// Weight transpose: dst[k*G + g] = src[g*srcLd + colOff + k], k in [0,K)
// ---------------------------------------------------------------------------
__global__ void fg_transpose(const float* __restrict__ src, float* __restrict__ dst,
                             int G, int srcLd, int colOff, int K) {
    int id = blockIdx.x * blockDim.x + threadIdx.x;
    if (id >= G * K) return;
    int g = id / K;
    int k = id - g * K;
    dst[(size_t)k * G + g] = src[(size_t)g * srcLd + colOff + k];
}

// ---------------------------------------------------------------------------
// field_pos[b,s,:] = wfe[fields] + wpe[pos] + wpe[rpos]  (written to output)
// ---------------------------------------------------------------------------
__global__ void fg_fieldpos(const int* __restrict__ fields, const int* __restrict__ pos,
                            const int* __restrict__ rpos,
                            const float* __restrict__ wfe, const float* __restrict__ wpe,
                            float* __restrict__ out_fp) {
    int bs  = blockIdx.x;          // b*S + s
    int tid = threadIdx.x;         // 0..127
    int f = fields[bs], p = pos[bs], r = rpos[bs];
    const float4* wfe4 = (const float4*)wfe;
    const float4* wpe4 = (const float4*)wpe;
    float4 a = wfe4[(size_t)f * 128 + tid];
    float4 b = wpe4[(size_t)p * 128 + tid];
    float4 c = wpe4[(size_t)r * 128 + tid];
    float4 o;
    o.x = a.x + b.x + c.x; o.y = a.y + b.y + c.y;
    o.z = a.z + b.z + c.z; o.w = a.w + b.w + c.w;
    ((float4*)out_fp)[(size_t)bs * 128 + tid] = o;
}

// ---------------------------------------------------------------------------
// gx GEMM:  gx[row, n] = sum_k wte[tok(row), k] * WxT[k, n] + bias[n]
// ---------------------------------------------------------------------------
__global__ void __launch_bounds__(128) fg_gemm_gx(
    const int* __restrict__ inputs, const float* __restrict__ wte,
    const float* __restrict__ WxT, const float* __restrict__ bias,
    float* __restrict__ out) {
    __shared__ __align__(16) float Bs[2 * BUFWORDS];
    const int tid  = threadIdx.x;
    const int wave = tid >> 5;
    const int lane = tid & 31;
    const int half = lane >> 4;
    const int l16  = lane & 15;
    const int nb = blockIdx.x, mb = blockIdx.y;

    int arow = mb * 64 + wave * 16 + l16;
    int bidx = arow & (Bsz - 1);
    int sidx = arow >> 6;
    int tok  = inputs[bidx * Ssz + sidx];
    const float* Ap = wte + (size_t)tok * Hsz;
    __builtin_prefetch(Ap, 0, 1);

    v8f acc[4] = {};
    fg_mm_pipe(Ap, WxT, G4, nb * 64, 16, Bs, tid, acc);

    int rowBase = mb * 64 + wave * 16 + 8 * half;
#pragma unroll
    for (int t = 0; t < 4; ++t) {
        int col = nb * 64 + t * 16 + l16;
        float bv = bias[col];
#pragma unroll
        for (int v = 0; v < 8; ++v)
            out[(size_t)(rowBase + v) * G4 + col] = acc[t][v] + bv;
    }
}

// ---------------------------------------------------------------------------
// zf GEMM + activation:  lz[row, n] = act(sum_k fp[row, k] * WfT[k, n] + bf[n])
// ---------------------------------------------------------------------------
__global__ void __launch_bounds__(128) fg_gemm_zf(
    const float* __restrict__ fp, const float* __restrict__ WfT,
    const float* __restrict__ bias, float* __restrict__ out) {
    __shared__ __align__(16) float Bs[2 * BUFWORDS];
    const int tid  = threadIdx.x;
    const int wave = tid >> 5;
    const int lane = tid & 31;
    const int half = lane >> 4;
    const int l16  = lane & 15;
    const int nb = blockIdx.x, mb = blockIdx.y;

    int arow = mb * 64 + wave * 16 + l16;
    int bidx = arow & (Bsz - 1);
    int sidx = arow >> 6;
    const float* Ap = fp + ((size_t)bidx * Ssz + sidx) * Hsz;
    __builtin_prefetch(Ap, 0, 1);

    v8f acc[4] = {};
    fg_mm_pipe(Ap, WfT, G2, nb * 64, 16, Bs, tid, acc);

    int rowBase = mb * 64 + wave * 16 + 8 * half;
#pragma unroll
    for (int t = 0; t < 4; ++t) {
        int col = nb * 64 + t * 16 + l16;
        float bv = bias[col];
        bool isSig = (col < Hsz);   // uniform per tile
#pragma unroll
        for (int v = 0; v < 8; ++v) {
            float g = acc[t][v] + bv;
            out[(size_t)(rowBase + v) * G2 + col] = isSig ? sigmf(g) : tanhf(g);
        }
    }
}

// ---------------------------------------------------------------------------
// One LSTM step:  g = gx_t + h_prev @ WhT ;  gates -> c,h update.
// A wave owns rows [w*16,w*16+16) x cols [jb*16,jb*16+16) across ALL FOUR
// gates (colStride = 512), so the state update is register-local.
// ---------------------------------------------------------------------------
__global__ void __launch_bounds__(128) fg_lstm_step(
    const float* __restrict__ hprev, float* __restrict__ hnext,
    float* __restrict__ cbuf,
    const float* __restrict__ gx_t,   // [64, 2048]
    const float* __restrict__ lz_t,   // [64, 1024]
    const float* __restrict__ WhT,    // [512, 2048]
    float* __restrict__ hs_t) {       // [64, 512] slice of hs output
    __shared__ __align__(16) float Bs[2 * BUFWORDS];
    const int tid  = threadIdx.x;
    const int wave = tid >> 5;
    const int lane = tid & 31;
    const int half = lane >> 4;
    const int l16  = lane & 15;
    const int jb = blockIdx.x;

    const float* Ap = hprev + (size_t)(wave * 16 + l16) * Hsz;
    __builtin_prefetch(gx_t + (size_t)(wave * 16) * G4 + jb * 16, 0, 1);
    __builtin_prefetch(lz_t + (size_t)(wave * 16) * G2 + jb * 16, 0, 1);

    v8f acc[4] = {};
    fg_mm_pipe(Ap, WhT, G4, jb * 16, Hsz, Bs, tid, acc);

    int rowb = wave * 16 + 8 * half;
    int j = jb * 16 + l16;
#pragma unroll
    for (int v = 0; v < 8; ++v) {
        int r = rowb + v;
        const float* gr = gx_t + (size_t)r * G4;
        float g0 = acc[0][v] + gr[j];
        float g1 = acc[1][v] + gr[Hsz + j];
        float g2 = acc[2][v] + gr[2 * Hsz + j];
        float g3 = acc[3][v] + gr[3 * Hsz + j];
        float i_ = sigmf(g0);
        float f_ = sigmf(g1);
        float o_ = sigmf(g2);
        float ch = tanhf(g3);
        float l_ = lz_t[(size_t)r * G2 + j];
        float z_ = lz_t[(size_t)r * G2 + Hsz + j];
        float cp = cbuf[r * Hsz + j];
        float cn = f_ * cp + i_ * ch + l_ * z_;
        cbuf[r * Hsz + j] = cn;
        float hn = o_ * tanhf(cn);
        hnext[r * Hsz + j] = hn;
        hs_t[r * Hsz + j]  = hn;
    }
}

// ---------------------------------------------------------------------------
__global__ void fg_zero(float* p, int n) {
    int id = blockIdx.x * blockDim.x + threadIdx.x;
    if (id < n) p[id] = 0.0f;
}

__global__ void fg_finalize(const float* __restrict__ hfin, const float* __restrict__ cfin,
                            float* __restrict__ out_h, float* __restrict__ out_c, int n) {
    int id = blockIdx.x * blockDim.x + threadIdx.x;
    if (id < n) { out_h[id] = hfin[id]; out_c[id] = cfin[id]; }
}

// ---------------------------------------------------------------------------
extern "C" void kernel_launch(void* const* d_in, const int* in_sizes, int n_in,
                              void* d_out, int out_size, void* d_ws, size_t ws_size,
                              hipStream_t stream) {
    (void)in_sizes; (void)n_in; (void)out_size; (void)ws_size;

    const int*   inputs = (const int*)d_in[0];
    const int*   fields = (const int*)d_in[1];
    const int*   pos    = (const int*)d_in[2];
    const int*   rpos   = (const int*)d_in[3];
    const float* wte    = (const float*)d_in[4];
    const float* wfe    = (const float*)d_in[5];
    const float* wpe    = (const float*)d_in[6];
    const float* w      = (const float*)d_in[7];
    const float* bvec   = (const float*)d_in[8];
    const float* wf     = (const float*)d_in[9];
    const float* bf     = (const float*)d_in[10];

    float* out = (float*)d_out;
    // output layout (floats): hs[S,B,H] | field_pos[B,S,H] | h[1,B,H] | c[1,B,H]
    float* out_hs = out;
    float* out_fp = out + (size_t)Ssz * Bsz * Hsz;
    float* out_h  = out_fp + (size_t)Bsz * Ssz * Hsz;
    float* out_c  = out_h + (size_t)Bsz * Hsz;

    // workspace layout (floats)
    float* ws   = (float*)d_ws;
    float* gx   = ws;                                 // 512*64*2048
    float* lz   = gx + (size_t)Ssz * Bsz * G4;        // 512*64*1024
    float* WxT  = lz + (size_t)Ssz * Bsz * G2;        // 512*2048
    float* WhT  = WxT + (size_t)Hsz * G4;             // 512*2048
    float* WfT  = WhT + (size_t)Hsz * G4;             // 512*1024
    float* h0   = WfT + (size_t)Hsz * G2;             // 64*512
    float* h1   = h0 + (size_t)Bsz * Hsz;
    float* cbuf = h1 + (size_t)Bsz * Hsz;

    // 1) transpose weights: WxT[k,g]=w[g,k]  WhT[k,g]=w[g,512+k]  WfT[k,g]=wf[g,k]
    fg_transpose<<<(G4 * Hsz) / 256, 256, 0, stream>>>(w,  WxT, G4, G2, 0,   Hsz);
    fg_transpose<<<(G4 * Hsz) / 256, 256, 0, stream>>>(w,  WhT, G4, G2, Hsz, Hsz);
    fg_transpose<<<(G2 * Hsz) / 256, 256, 0, stream>>>(wf, WfT, G2, Hsz, 0,  Hsz);

    // 2) field_pos gather -> output slot 2 (also the A operand of the zf GEMM)
    fg_fieldpos<<<Bsz * Ssz, 128, 0, stream>>>(fields, pos, rpos, wfe, wpe, out_fp);

    // 3) big time-parallel GEMMs
    {
        dim3 grid(G4 / 64, (Ssz * Bsz) / 64);   // 32 x 512
        fg_gemm_gx<<<grid, 128, 0, stream>>>(inputs, wte, WxT, bvec, gx);
    }
    {
        dim3 grid(G2 / 64, (Ssz * Bsz) / 64);   // 16 x 512
        fg_gemm_zf<<<grid, 128, 0, stream>>>(out_fp, WfT, bf, lz);
    }

    // 4) h0 = 0, c0 = 0
    fg_zero<<<(3 * Bsz * Hsz + 255) / 256, 256, 0, stream>>>(h0, 3 * Bsz * Hsz);

    // 5) sequential scan: 512 dependent micro-GEMM steps (grid sync = launch boundary)
    float* hbuf[2] = {h0, h1};
    for (int t = 0; t < Ssz; ++t) {
        fg_lstm_step<<<Hsz / 16, 128, 0, stream>>>(
            hbuf[t & 1], hbuf[(t + 1) & 1], cbuf,
            gx + (size_t)t * Bsz * G4,
            lz + (size_t)t * Bsz * G2,
            WhT,
            out_hs + (size_t)t * Bsz * Hsz);
    }

    // 6) final h (= hbuf[0] after 512 steps) and c
    fg_finalize<<<(Bsz * Hsz + 255) / 256, 256, 0, stream>>>(hbuf[0], cbuf, out_h, out_c,
                                                             Bsz * Hsz);
}